// CrystalHypergraphConv_73435350827165
// MI455X (gfx1250) — compile-verified
//
#include <hip/hip_runtime.h>
#include <math.h>

typedef __attribute__((ext_vector_type(2))) float v2f;
typedef __attribute__((ext_vector_type(8))) float v8f;

#define NUM_GRAPHS 128
#define BN_EPS 1e-5f

// ---------------- math helpers ----------------
__device__ __forceinline__ float softplusf(float x) {
    // log1p(exp(x)), stable: max(x,0) + log1p(exp(-|x|))
    return fmaxf(x, 0.0f) + log1pf(expf(-fabsf(x)));
}
__device__ __forceinline__ float sigmoidf(float x) {
    return 1.0f / (1.0f + expf(-x));
}
// Ordered float atomic max via signed-max / unsigned-min trick (init must be -inf).
__device__ __forceinline__ void atomicMaxFloat(float* addr, float val) {
    if (val >= 0.0f)
        atomicMax((int*)addr, __float_as_int(val));
    else
        atomicMin((unsigned int*)addr, (unsigned int)__float_as_int(val));
}

// ---------------- utility kernels ----------------
__global__ void fill_kernel(float* __restrict__ p, float v, int n) {
    int i = blockIdx.x * blockDim.x + threadIdx.x;
    int stride = gridDim.x * blockDim.x;
    for (; i < n; i += stride) p[i] = v;
}

// Wsum[r,c] = lin_w[r,c] + lin_w[256+r,c]   (r<128, c<256)
__global__ void wsum_kernel(const float* __restrict__ lin_w, float* __restrict__ Wsum) {
    int i = blockIdx.x * blockDim.x + threadIdx.x;
    if (i < 128 * 256) {
        int r = i >> 8, c = i & 255;
        Wsum[i] = lin_w[r * 256 + c] + lin_w[(256 + r) * 256 + c];
    }
}

// ---------------- WMMA f32 GEMM: C[M,Nc] = A[M,K] @ W[K,Nc] (+bias)(+softplus) ----------------
// One wave32 computes one 16x16 tile of C using V_WMMA_F32_16X16X4_F32.
// Requires M%16==0, Nc%16==0, K%4==0 (true for all call sites).
__global__ void gemm_wmma_f32(const float* __restrict__ A, const float* __restrict__ Wt,
                              const float* __restrict__ bias, float* __restrict__ C,
                              int M, int K, int Nc, int act) {
    int wave = (blockIdx.x * blockDim.x + threadIdx.x) >> 5;
    int lane = threadIdx.x & 31;
    int tilesN = Nc >> 4;
    int tileM = wave / tilesN;
    int tileN = wave - tileM * tilesN;
    if (tileM * 16 >= M) return;   // wave-uniform: EXEC stays all-ones for WMMA

    int rowA  = tileM * 16 + (lane & 15);  // A: lanes 0-15 and 16-31 both map M=0..15
    int colB  = tileN * 16 + (lane & 15);  // B/D: lane%16 = column
    int khalf = (lane >> 4) * 2;           // lanes 16-31 hold K+2,K+3

    v8f acc = {};
    for (int k0 = 0; k0 < K; k0 += 4) {
        v2f a, b;
        a.x = A[(size_t)rowA * K + k0 + khalf];
        a.y = A[(size_t)rowA * K + k0 + khalf + 1];
        b.x = Wt[(size_t)(k0 + khalf) * Nc + colB];
        b.y = Wt[(size_t)(k0 + khalf + 1) * Nc + colB];
        acc = __builtin_amdgcn_wmma_f32_16x16x4_f32(
            /*neg_a=*/false, a, /*neg_b=*/false, b,
            /*c_mod=*/(short)0, acc, /*reuse_a=*/false, /*reuse_b=*/false);
    }

    // D layout: VGPR v, lanes 0-15 -> M=v ; lanes 16-31 -> M=v+8 ; N = lane%16
    int mbase = tileM * 16 + ((lane >> 4) << 3);
    float bv = bias ? bias[colB] : 0.0f;
#pragma unroll
    for (int v = 0; v < 8; ++v) {
        float val = acc[v] + bv;
        if (act) val = softplusf(val);
        C[(size_t)(mbase + v) * Nc + colB] = val;
    }
}

// ---------------- edge pass 1: BN1 statistics over z = P[idx0] + Q[idx1] ----------------
__global__ void bn1_stats_kernel(const float* __restrict__ P, const float* __restrict__ Q,
                                 const int* __restrict__ idx0, const int* __restrict__ idx1,
                                 int E, float* __restrict__ sums, float* __restrict__ sumsq) {
    int c = threadIdx.x;  // 0..255
    float s = 0.0f, s2 = 0.0f;
    for (int e = blockIdx.x; e < E; e += gridDim.x) {
        int i0 = idx0[e], i1 = idx1[e];
        float z = P[(size_t)i0 * 256 + c] + Q[(size_t)i1 * 256 + c];
        s += z;
        s2 += z * z;
    }
    atomicAdd(&sums[c], s);
    atomicAdd(&sumsq[c], s2);
}

// BN finalize: scale/shift from sums (works for bn1 C=256 and bn2 C=128)
__global__ void bn_finalize_kernel(const float* __restrict__ sums, const float* __restrict__ sumsq,
                                   const float* __restrict__ g, const float* __restrict__ b,
                                   float cntInv, int C,
                                   float* __restrict__ scale, float* __restrict__ shift) {
    int c = blockIdx.x * blockDim.x + threadIdx.x;
    if (c < C) {
        float mean = sums[c] * cntInv;
        float var  = sumsq[c] * cntInv - mean * mean;
        float sc   = g[c] * rsqrtf(var + BN_EPS);
        scale[c] = sc;
        shift[c] = b[c] - mean * sc;
    }
}

// ---------------- edge pass 2: per-(node,channel) segment max of alpha ----------------
__global__ void segmax_kernel(const float* __restrict__ P, const float* __restrict__ Q,
                              const int* __restrict__ idx0, const int* __restrict__ idx1,
                              const float* __restrict__ scale, const float* __restrict__ shift,
                              const float* __restrict__ aggr_t, int E, float* __restrict__ m) {
    int c = threadIdx.x;  // 0..127
    float t   = aggr_t[0];
    float scf = scale[c],       shf = shift[c];
    float scc = scale[c + 128], shc = shift[c + 128];
    for (int e = blockIdx.x; e < E; e += gridDim.x) {
        int i0 = idx0[e], i1 = idx1[e];
        float zf = P[(size_t)i0 * 256 + c]       + Q[(size_t)i1 * 256 + c];
        float zc = P[(size_t)i0 * 256 + 128 + c] + Q[(size_t)i1 * 256 + 128 + c];
        float msg = sigmoidf(zf * scf + shf) * softplusf(zc * scc + shc);
        atomicMaxFloat(&m[(size_t)i0 * 128 + c], msg * t);
    }
}

// ---------------- edge pass 3: exp-weighted segment sums ----------------
__global__ void aggregate_kernel(const float* __restrict__ P, const float* __restrict__ Q,
                                 const int* __restrict__ idx0, const int* __restrict__ idx1,
                                 const float* __restrict__ scale, const float* __restrict__ shift,
                                 const float* __restrict__ aggr_t, const float* __restrict__ m,
                                 int E, float* __restrict__ denom, float* __restrict__ num) {
    int c = threadIdx.x;  // 0..127
    float t   = aggr_t[0];
    float scf = scale[c],       shf = shift[c];
    float scc = scale[c + 128], shc = shift[c + 128];
    for (int e = blockIdx.x; e < E; e += gridDim.x) {
        int i0 = idx0[e], i1 = idx1[e];
        float zf = P[(size_t)i0 * 256 + c]       + Q[(size_t)i1 * 256 + c];
        float zc = P[(size_t)i0 * 256 + 128 + c] + Q[(size_t)i1 * 256 + 128 + c];
        float msg = sigmoidf(zf * scf + shf) * softplusf(zc * scc + shc);
        float ex  = expf(msg * t - m[(size_t)i0 * 128 + c]);
        atomicAdd(&denom[(size_t)i0 * 128 + c], ex);
        atomicAdd(&num[(size_t)i0 * 128 + c], msg * ex);
    }
}

// out = num/(denom+1e-16) (in place into num) + accumulate BN2 stats
__global__ void node_out_stats_kernel(float* __restrict__ num, const float* __restrict__ denom,
                                      int N, float* __restrict__ sums, float* __restrict__ sumsq) {
    int c = threadIdx.x;  // 0..127
    float s = 0.0f, s2 = 0.0f;
    for (int n = blockIdx.x; n < N; n += gridDim.x) {
        size_t i = (size_t)n * 128 + c;
        float o = num[i] / (denom[i] + 1e-16f);
        num[i] = o;
        s += o;
        s2 += o * o;
    }
    atomicAdd(&sums[c], s);
    atomicAdd(&sumsq[c], s2);
}

// h2 = softplus(bn2(out) + h); pooled[batch] += h2 ; counts[batch] += 1
__global__ void residual_pool_kernel(const float* __restrict__ out, const float* __restrict__ h,
                                     const float* __restrict__ scale, const float* __restrict__ shift,
                                     const int* __restrict__ batch, int N,
                                     float* __restrict__ pooled, float* __restrict__ counts) {
    int c = threadIdx.x;  // 0..127
    float sc = scale[c], sh = shift[c];
    for (int n = blockIdx.x; n < N; n += gridDim.x) {
        size_t i = (size_t)n * 128 + c;
        float v = softplusf(out[i] * sc + sh + h[i]);
        int g = batch[n];
        atomicAdd(&pooled[(size_t)g * 128 + c], v);
        if (c == 0) atomicAdd(&counts[g], 1.0f);
    }
}

__global__ void pool_div_kernel(float* __restrict__ pooled, const float* __restrict__ counts) {
    int i = blockIdx.x * blockDim.x + threadIdx.x;  // 128*128
    if (i < NUM_GRAPHS * 128) pooled[i] /= fmaxf(counts[i >> 7], 1.0f);
}

// final: d_out[g] = dot(g1[g,:], out_w) + out_b
__global__ void head_kernel(const float* __restrict__ g1, const float* __restrict__ out_w,
                            const float* __restrict__ out_b, float* __restrict__ out) {
    int g = blockIdx.x * blockDim.x + threadIdx.x;
    if (g < NUM_GRAPHS) {
        float acc = 0.0f;
        for (int c = 0; c < 256; ++c) acc += g1[(size_t)g * 256 + c] * out_w[c];
        out[g] = acc + out_b[0];
    }
}

// ---------------- launch ----------------
extern "C" void kernel_launch(void* const* d_in, const int* in_sizes, int n_in,
                              void* d_out, int out_size, void* d_ws, size_t ws_size,
                              hipStream_t stream) {
    const float* x        = (const float*)d_in[0];
    const float* hedge    = (const float*)d_in[1];
    const int*   rel      = (const int*)d_in[2];
    const int*   batch    = (const int*)d_in[3];
    const float* embed_w  = (const float*)d_in[5];
    const float* embed_b  = (const float*)d_in[6];
    const float* bembed_w = (const float*)d_in[7];
    const float* bembed_b = (const float*)d_in[8];
    const float* lin_w    = (const float*)d_in[9];
    const float* lin_b    = (const float*)d_in[10];
    const float* bn1_g    = (const float*)d_in[11];
    const float* bn1_b    = (const float*)d_in[12];
    const float* bn2_g    = (const float*)d_in[13];
    const float* bn2_b    = (const float*)d_in[14];
    const float* aggr_t   = (const float*)d_in[15];
    const float* l1_w     = (const float*)d_in[16];
    const float* l1_b     = (const float*)d_in[17];
    const float* out_w    = (const float*)d_in[18];
    const float* out_b    = (const float*)d_in[19];

    const int N   = in_sizes[0] / 92;
    const int NHE = in_sizes[1] / 40;
    const int E   = in_sizes[2] / 3;
    const int* idx0 = rel;
    const int* idx1 = rel + E;
    const float* Wconn = lin_w + 128 * 256;  // rows 128..255 of lin_w

    // ---- workspace layout ----
    float* W = (float*)d_ws;
    size_t off = 0;
    float* h    = W + off; off += (size_t)N * 128;
    float* cf   = W + off; off += (size_t)NHE * 128;
    float* P    = W + off; off += (size_t)N * 256;
    float* Q    = W + off; off += (size_t)NHE * 256;
    float* Wsum = W + off; off += 128 * 256;
    // zero-init region (one fill):
    float* zstart = W + off;
    float* denom = W + off; off += (size_t)N * 128;
    float* num   = W + off; off += (size_t)N * 128;   // becomes `out` in place
    float* sums1 = W + off; off += 256;
    float* sumq1 = W + off; off += 256;
    float* scl1  = W + off; off += 256;
    float* shf1  = W + off; off += 256;
    float* sums2 = W + off; off += 128;
    float* sumq2 = W + off; off += 128;
    float* scl2  = W + off; off += 128;
    float* shf2  = W + off; off += 128;
    float* pooled= W + off; off += (size_t)NUM_GRAPHS * 128;
    float* counts= W + off; off += NUM_GRAPHS;
    int zcount = (int)((W + off) - zstart);
    // -inf region:
    float* m = W + off; off += (size_t)N * 128;
    float* g1 = W + off; off += (size_t)NUM_GRAPHS * 256;

    // ---- init ----
    fill_kernel<<<1024, 256, 0, stream>>>(zstart, 0.0f, zcount);
    fill_kernel<<<1024, 256, 0, stream>>>(m, -INFINITY, N * 128);
    wsum_kernel<<<(128 * 256 + 255) / 256, 256, 0, stream>>>(lin_w, Wsum);

    // ---- node GEMMs (WMMA f32) ----
    auto gemm = [&](const float* A, const float* Wt, const float* bias, float* C,
                    int M, int K, int Nc, int act) {
        int tiles = (M / 16) * (Nc / 16);
        int blocks = (tiles + 3) / 4;  // 4 waves / 128-thread block
        gemm_wmma_f32<<<blocks, 128, 0, stream>>>(A, Wt, bias, C, M, K, Nc, act);
    };
    gemm(x,      embed_w,  embed_b,  h,  N,   92,  128, 0);  // h  = x @ embed_w + b
    gemm(hedge,  bembed_w, bembed_b, cf, NHE, 40,  128, 0);  // cf = hedge @ bembed_w + b
    gemm(h,      Wsum,     lin_b,    P,  N,   128, 256, 0);  // P  = h @ (Wa+Wc) + lin_b
    gemm(cf,     Wconn,    nullptr,  Q,  NHE, 128, 256, 0);  // Q  = cf @ Wb

    // ---- edge pass 1: BN1 stats ----
    int gb = E < 4096 ? E : 4096;
    bn1_stats_kernel<<<gb, 256, 0, stream>>>(P, Q, idx0, idx1, E, sums1, sumq1);
    bn_finalize_kernel<<<1, 256, 0, stream>>>(sums1, sumq1, bn1_g, bn1_b,
                                              1.0f / (float)E, 256, scl1, shf1);

    // ---- edge pass 2: segment max ----
    int eb = E < 8192 ? E : 8192;
    segmax_kernel<<<eb, 128, 0, stream>>>(P, Q, idx0, idx1, scl1, shf1, aggr_t, E, m);

    // ---- edge pass 3: softmax-weighted aggregation ----
    aggregate_kernel<<<eb, 128, 0, stream>>>(P, Q, idx0, idx1, scl1, shf1, aggr_t, m,
                                             E, denom, num);

    // ---- node finalize: out = num/(denom+eps), BN2 stats ----
    int nb = N < 4096 ? N : 4096;
    node_out_stats_kernel<<<nb, 128, 0, stream>>>(num, denom, N, sums2, sumq2);
    bn_finalize_kernel<<<1, 128, 0, stream>>>(sums2, sumq2, bn2_g, bn2_b,
                                              1.0f / (float)N, 128, scl2, shf2);

    // ---- residual + softplus + mean pool ----
    residual_pool_kernel<<<nb, 128, 0, stream>>>(num, h, scl2, shf2, batch, N, pooled, counts);
    pool_div_kernel<<<(NUM_GRAPHS * 128 + 255) / 256, 256, 0, stream>>>(pooled, counts);

    // ---- head: g1 = softplus(pooled @ l1_w + l1_b), out = g1 @ out_w + out_b ----
    gemm(pooled, l1_w, l1_b, g1, NUM_GRAPHS, 128, 256, 1);
    head_kernel<<<1, 128, 0, stream>>>(g1, out_w, out_b, (float*)d_out);
}